// RGAT_82197084110912
// MI455X (gfx1250) — compile-verified
//
#include <hip/hip_runtime.h>
#include <hip/hip_bf16.h>
#include <math.h>

// ---------------- problem constants (match reference) ----------------
#define N_NODES   50000
#define N_EDGES   800000
#define FEATS     128          // in/hidden width (both 128)
#define OUT_FEATS 64
#define ROW_TILES (N_NODES / 16)   // 3125 exact
#define COS_EPS   1e-8f
#define LN_EPS    1e-5f
#define THRESH    0.1f

typedef float v2f __attribute__((ext_vector_type(2)));
typedef float v8f __attribute__((ext_vector_type(8)));

// ---------------- helpers ----------------
__device__ __forceinline__ void atomicMaxF32(float* addr, float val) {
    unsigned int* ua = (unsigned int*)addr;
    unsigned int old = *ua;
    while (__uint_as_float(old) < val) {
        unsigned int assumed = old;
        old = atomicCAS(ua, assumed, __float_as_uint(val));
        if (old == assumed) break;
    }
}

// ---------------- GEMM: C[N x ncols] = X[N x 128] @ W[128 x ncols] ----------------
// f32 WMMA 16x16x4. One block = one 16-row tile; wave w handles columns [16w,16w+16).
// A tile (16x128 f32, 8KB) staged in LDS once per block, shared by all waves.
__global__ void gemm_wmma_f32(const float* __restrict__ X,
                              const float* __restrict__ W,
                              float* __restrict__ C,
                              int ncols) {
    __shared__ float sx[16 * FEATS];

    const int rowTile = blockIdx.x;
    // cooperative, coalesced fill of the A tile
    for (int i = threadIdx.x; i < 16 * FEATS; i += blockDim.x)
        sx[i] = X[(size_t)rowTile * (16 * FEATS) + i];
    __syncthreads();

    const int wave = threadIdx.x >> 5;
    const int lane = threadIdx.x & 31;
    const int half = lane >> 4;     // 0: lanes 0-15, 1: lanes 16-31
    const int mn   = lane & 15;     // M for A / N for B
    const int col0 = wave * 16;

    v8f acc = {};
    for (int k0 = 0; k0 < FEATS; k0 += 4) {
        const int k = k0 + half * 2;            // half-wave K split (ISA 16x4 layout)
        v2f a, b;
        a.x = sx[mn * FEATS + k];               // A: row mn, K = k, k+1
        a.y = sx[mn * FEATS + k + 1];
        b.x = W[(size_t)k       * ncols + col0 + mn];   // B: rows k,k+1, col mn
        b.y = W[(size_t)(k + 1) * ncols + col0 + mn];
        acc = __builtin_amdgcn_wmma_f32_16x16x4_f32(
                  false, a, false, b, (short)0, acc, false, false);
    }

    // C/D layout: VGPR v -> row v (lanes 0-15) / row v+8 (lanes 16-31), col = lane&15
    const int mbase = rowTile * 16 + half * 8;
    #pragma unroll
    for (int v = 0; v < 8; ++v)
        C[(size_t)(mbase + v) * ncols + col0 + mn] = acc[v];
}

// ---------------- per-node attention logit: e = log(clamped cosine) ----------------
__global__ void node_score_kernel(const float* __restrict__ ft,
                                  float* __restrict__ e_node, int D) {
    const int node = (blockIdx.x * blockDim.x + threadIdx.x) >> 5;
    const int lane = threadIdx.x & 31;
    if (node >= N_NODES) return;
    const float* row = ft + (size_t)node * D;
    float sq = 0.f;
    for (int i = lane; i < D; i += 32) { float v = row[i]; sq += v * v; }
    #pragma unroll
    for (int off = 16; off; off >>= 1) sq += __shfl_xor(sq, off, 32);
    if (lane == 0) {
        float nrm  = fmaxf(sqrtf(sq), COS_EPS);
        float cosv = sq / (nrm * nrm);
        float c    = (cosv >= THRESH) ? cosv : 1e-6f;
        e_node[node] = logf(c);
    }
}

__global__ void fill_f32(float* __restrict__ p, float v, int n) {
    int i = blockIdx.x * blockDim.x + threadIdx.x;
    if (i < n) p[i] = v;
}

// ---------------- edge segment-max over dst ----------------
__global__ void edge_max_kernel(const int* __restrict__ src, const int* __restrict__ dst,
                                const float* __restrict__ e_node, float* __restrict__ m) {
    int e = blockIdx.x * blockDim.x + threadIdx.x;
    if (e >= N_EDGES) return;
    atomicMaxF32(&m[dst[e]], e_node[src[e]]);
}

// ---------------- edge exp + segment-sum ----------------
__global__ void edge_expsum_kernel(const int* __restrict__ src, const int* __restrict__ dst,
                                   const float* __restrict__ e_node, const float* __restrict__ m,
                                   float* __restrict__ ex, float* __restrict__ den) {
    int e = blockIdx.x * blockDim.x + threadIdx.x;
    if (e >= N_EDGES) return;
    float v = expf(e_node[src[e]] - m[dst[e]]);
    ex[e] = v;
    atomicAdd(&den[dst[e]], v);
}

// ---------------- weighted aggregation: agg[dst] += ft[src] * a  (wave per edge) ----
__global__ void edge_agg_kernel(const int* __restrict__ src, const int* __restrict__ dst,
                                const float* __restrict__ ft, const float* __restrict__ ex,
                                const float* __restrict__ den, float* __restrict__ agg, int D) {
    const int e    = (blockIdx.x * blockDim.x + threadIdx.x) >> 5;
    const int lane = threadIdx.x & 31;
    if (e >= N_EDGES) return;
    const int s = src[e], d = dst[e];
    const float a = ex[e] / fmaxf(den[d], 1e-16f);
    const float* fs = ft  + (size_t)s * D;
    float*       od = agg + (size_t)d * D;
    for (int i = lane; i < D; i += 32)
        atomicAdd(&od[i], fs[i] * a);
}

// ---------------- bias + LayerNorm + ReLU (wave per node, D=128) ----------------
__global__ void ln_relu_kernel(const float* __restrict__ agg, const float* __restrict__ bias,
                               const float* __restrict__ g, const float* __restrict__ beta,
                               float* __restrict__ out) {
    const int node = (blockIdx.x * blockDim.x + threadIdx.x) >> 5;
    const int lane = threadIdx.x & 31;
    if (node >= N_NODES) return;
    const float* row = agg + (size_t)node * FEATS;
    float v[4];
    float s = 0.f;
    #pragma unroll
    for (int j = 0; j < 4; ++j) {
        int i = lane + j * 32;
        v[j] = row[i] + bias[i];
        s += v[j];
    }
    #pragma unroll
    for (int off = 16; off; off >>= 1) s += __shfl_xor(s, off, 32);
    const float mu = s * (1.f / FEATS);
    float vs = 0.f;
    #pragma unroll
    for (int j = 0; j < 4; ++j) { float d = v[j] - mu; vs += d * d; }
    #pragma unroll
    for (int off = 16; off; off >>= 1) vs += __shfl_xor(vs, off, 32);
    const float inv = rsqrtf(vs * (1.f / FEATS) + LN_EPS);
    float* orow = out + (size_t)node * FEATS;
    #pragma unroll
    for (int j = 0; j < 4; ++j) {
        int i = lane + j * 32;
        float y = (v[j] - mu) * inv * g[i] + beta[i];
        orow[i] = fmaxf(y, 0.f);
    }
}

// ---------------- final bias add (layer 3 output) ----------------
__global__ void bias_add_kernel(const float* __restrict__ agg, const float* __restrict__ b,
                                float* __restrict__ out) {
    int i = blockIdx.x * blockDim.x + threadIdx.x;
    if (i < N_NODES * OUT_FEATS) out[i] = agg[i] + b[i & (OUT_FEATS - 1)];
}

// ---------------- orchestration ----------------
extern "C" void kernel_launch(void* const* d_in, const int* in_sizes, int n_in,
                              void* d_out, int out_size, void* d_ws, size_t ws_size,
                              hipStream_t stream) {
    (void)in_sizes; (void)n_in; (void)out_size; (void)ws_size;

    const float* feat = (const float*)d_in[0];
    const int*   src  = (const int*)d_in[1];
    const int*   dst  = (const int*)d_in[2];
    const float* W0   = (const float*)d_in[3];
    const float* b0   = (const float*)d_in[4];
    const float* W1   = (const float*)d_in[5];
    const float* b1   = (const float*)d_in[6];
    const float* W2   = (const float*)d_in[7];
    const float* b2   = (const float*)d_in[8];
    const float* ln1g = (const float*)d_in[9];
    const float* ln1b = (const float*)d_in[10];
    const float* ln2g = (const float*)d_in[11];
    const float* ln2b = (const float*)d_in[12];

    // workspace layout (256B aligned slices)
    char* w = (char*)d_ws;
    size_t off = 0;
    auto take = [&](size_t bytes) {
        void* p = w + off;
        off = (off + bytes + 255) & ~(size_t)255;
        return p;
    };
    float* x    = (float*)take((size_t)N_NODES * FEATS * 4); // activations between layers
    float* ft   = (float*)take((size_t)N_NODES * FEATS * 4); // GEMM output
    float* agg  = (float*)take((size_t)N_NODES * FEATS * 4); // segment-sum accumulator
    float* en   = (float*)take((size_t)N_NODES * 4);         // per-node logits
    float* m    = (float*)take((size_t)N_NODES * 4);         // segment max
    float* den  = (float*)take((size_t)N_NODES * 4);         // segment sum of exp
    float* ex   = (float*)take((size_t)N_EDGES * 4);         // per-edge exp values

    const int tEdges = (N_EDGES + 255) / 256;                 // thread-per-edge grids
    const int wEdges = (N_EDGES * 32 + 255) / 256;            // wave-per-edge grids
    const int wNodes = (N_NODES * 32 + 255) / 256;            // wave-per-node grids

    auto run_conv = [&](const float* Xin, const float* Wm, int D, float* aggOut) {
        gemm_wmma_f32<<<ROW_TILES, (D / 16) * 32, 0, stream>>>(Xin, Wm, ft, D);
        node_score_kernel<<<wNodes, 256, 0, stream>>>(ft, en, D);
        fill_f32<<<(N_NODES + 255) / 256, 256, 0, stream>>>(m, -INFINITY, N_NODES);
        fill_f32<<<(N_NODES + 255) / 256, 256, 0, stream>>>(den, 0.f, N_NODES);
        fill_f32<<<(N_NODES * D + 255) / 256, 256, 0, stream>>>(aggOut, 0.f, N_NODES * D);
        edge_max_kernel<<<tEdges, 256, 0, stream>>>(src, dst, en, m);
        edge_expsum_kernel<<<tEdges, 256, 0, stream>>>(src, dst, en, m, ex, den);
        edge_agg_kernel<<<wEdges, 256, 0, stream>>>(src, dst, ft, ex, den, aggOut, D);
    };

    // layer 1: conv0 -> +b0 -> LN1 -> ReLU
    run_conv(feat, W0, FEATS, agg);
    ln_relu_kernel<<<wNodes, 256, 0, stream>>>(agg, b0, ln1g, ln1b, x);

    // layer 2: conv1 -> +b1 -> LN2 -> ReLU
    run_conv(x, W1, FEATS, agg);
    ln_relu_kernel<<<wNodes, 256, 0, stream>>>(agg, b1, ln2g, ln2b, x);

    // layer 3: conv2 (single head, 64 cols) -> +b2 -> output
    run_conv(x, W2, OUT_FEATS, agg);
    bias_add_kernel<<<(N_NODES * OUT_FEATS + 255) / 256, 256, 0, stream>>>(agg, b2, (float*)d_out);
}